// PEPair_78718160601617
// MI455X (gfx1250) — compile-verified
//
#include <hip/hip_runtime.h>

// Problem constants (from reference setup_inputs): B=4, N=1024, Din=64, Dout=64
// out[b,m,o] = max_n(x[b,n,:]·W1[o,:]) + x[b,m,:]·W2[o,:] + bias[o]

typedef __attribute__((ext_vector_type(2))) float v2f;
typedef __attribute__((ext_vector_type(8))) float v8f;

#define ROWS   4096   // B*N
#define DIN    64
#define DOUT   64
#define WSTR   128    // W row stride (2*Din)

// Monotonic order-preserving float <-> uint mapping (handles negatives).
__device__ __forceinline__ unsigned f32_to_ord(float f) {
  unsigned u = __float_as_uint(f);
  return (u & 0x80000000u) ? ~u : (u | 0x80000000u);
}
__device__ __forceinline__ float ord_to_f32(unsigned u) {
  return __uint_as_float((u & 0x80000000u) ? (u ^ 0x80000000u) : ~u);
}

// Max over the 16 rows of one 16x16 C/D tile, per column.
// C/D layout: VGPR v holds row (v + 8*(lane>=16)), col (lane&15).
__device__ __forceinline__ float tile_colmax(v8f a) {
  float m = a[0];
#pragma unroll
  for (int v = 1; v < 8; ++v) m = fmaxf(m, a[v]);
  m = fmaxf(m, __shfl_xor(m, 16, 32));   // combine rows 0-7 with rows 8-15
  return m;                               // valid in all lanes for col = lane&15
}

__global__ void pepair_init_amax(unsigned* __restrict__ amax) {
  amax[threadIdx.x] = 0u;                 // ord(-inf) = 0x007FFFFF > 0, so 0 is a safe floor
}

// Pass 1: a = X @ W1^T ; amax[b,o] = max over rows-in-batch via atomicMax(ord).
__global__ __launch_bounds__(256) void pepair_pass1(const float* __restrict__ x,
                                                    const float* __restrict__ W,
                                                    unsigned* __restrict__ amax) {
  const int lane  = threadIdx.x & 31;
  const int wave  = threadIdx.x >> 5;
  const int strip = blockIdx.x * 8 + wave;   // 16-row strip, 0..255
  const int row0  = strip << 4;
  const int batch = row0 >> 10;              // / N
  const int r     = lane & 15;
  const int kh    = (lane >> 4) << 1;        // 0 or 2: this lane's K sub-pair

  const float* xrow = x + (size_t)(row0 + r) * DIN + kh;   // A: row r of tile
  const float* wrow = W + (size_t)r * WSTR + kh;           // B: W1[col=r][k] (first half)

  v8f acc0 = {}, acc1 = {}, acc2 = {}, acc3 = {};
#pragma unroll
  for (int kk = 0; kk < DIN; kk += 4) {
    v2f a  = *(const v2f*)(xrow + kk);
    v2f b0 = *(const v2f*)(wrow + kk);
    v2f b1 = *(const v2f*)(wrow + 16 * WSTR + kk);
    v2f b2 = *(const v2f*)(wrow + 32 * WSTR + kk);
    v2f b3 = *(const v2f*)(wrow + 48 * WSTR + kk);
    acc0 = __builtin_amdgcn_wmma_f32_16x16x4_f32(false, a, false, b0, (short)0, acc0, false, false);
    acc1 = __builtin_amdgcn_wmma_f32_16x16x4_f32(false, a, false, b1, (short)0, acc1, false, false);
    acc2 = __builtin_amdgcn_wmma_f32_16x16x4_f32(false, a, false, b2, (short)0, acc2, false, false);
    acc3 = __builtin_amdgcn_wmma_f32_16x16x4_f32(false, a, false, b3, (short)0, acc3, false, false);
  }

  float c0 = tile_colmax(acc0);
  float c1 = tile_colmax(acc1);
  float c2 = tile_colmax(acc2);
  float c3 = tile_colmax(acc3);

  if (lane < 16) {
    unsigned* base = amax + batch * DOUT;
    atomicMax(base +  0 + r, f32_to_ord(c0));
    atomicMax(base + 16 + r, f32_to_ord(c1));
    atomicMax(base + 32 + r, f32_to_ord(c2));
    atomicMax(base + 48 + r, f32_to_ord(c3));
  }
}

// Pass 2: out = X @ W2^T + amax[batch,:] + bias
__global__ __launch_bounds__(256) void pepair_pass2(const float* __restrict__ x,
                                                    const float* __restrict__ W,
                                                    const float* __restrict__ bias,
                                                    const unsigned* __restrict__ amax,
                                                    float* __restrict__ out) {
  const int lane  = threadIdx.x & 31;
  const int wave  = threadIdx.x >> 5;
  const int strip = blockIdx.x * 8 + wave;
  const int row0  = strip << 4;
  const int batch = row0 >> 10;
  const int r     = lane & 15;
  const int kh    = (lane >> 4) << 1;
  const int hi    = lane >> 4;

  const float* xrow = x + (size_t)(row0 + r) * DIN + kh;
  const float* wrow = W + (size_t)r * WSTR + DIN + kh;     // second half of W -> W2

  v8f acc0 = {}, acc1 = {}, acc2 = {}, acc3 = {};
#pragma unroll
  for (int kk = 0; kk < DIN; kk += 4) {
    v2f a  = *(const v2f*)(xrow + kk);
    v2f b0 = *(const v2f*)(wrow + kk);
    v2f b1 = *(const v2f*)(wrow + 16 * WSTR + kk);
    v2f b2 = *(const v2f*)(wrow + 32 * WSTR + kk);
    v2f b3 = *(const v2f*)(wrow + 48 * WSTR + kk);
    acc0 = __builtin_amdgcn_wmma_f32_16x16x4_f32(false, a, false, b0, (short)0, acc0, false, false);
    acc1 = __builtin_amdgcn_wmma_f32_16x16x4_f32(false, a, false, b1, (short)0, acc1, false, false);
    acc2 = __builtin_amdgcn_wmma_f32_16x16x4_f32(false, a, false, b2, (short)0, acc2, false, false);
    ac3: ;
    acc3 = __builtin_amdgcn_wmma_f32_16x16x4_f32(false, a, false, b3, (short)0, acc3, false, false);
  }

  const unsigned* abase = amax + batch * DOUT;
  float add0 = ord_to_f32(abase[ 0 + r]) + bias[ 0 + r];
  float add1 = ord_to_f32(abase[16 + r]) + bias[16 + r];
  float add2 = ord_to_f32(abase[32 + r]) + bias[32 + r];
  float add3 = ord_to_f32(abase[48 + r]) + bias[48 + r];

#pragma unroll
  for (int v = 0; v < 8; ++v) {
    float* o = out + (size_t)(row0 + hi * 8 + v) * DOUT;
    o[ 0 + r] = acc0[v] + add0;
    o[16 + r] = acc1[v] + add1;
    o[32 + r] = acc2[v] + add2;
    o[48 + r] = acc3[v] + add3;
  }
}

extern "C" void kernel_launch(void* const* d_in, const int* in_sizes, int n_in,
                              void* d_out, int out_size, void* d_ws, size_t ws_size,
                              hipStream_t stream) {
  const float* x    = (const float*)d_in[0];   // (4,1024,64) f32
  const float* W    = (const float*)d_in[1];   // (64,128)    f32
  const float* bias = (const float*)d_in[2];   // (64,)       f32
  float*       out  = (float*)d_out;           // (4,1024,64) f32
  unsigned*    amax = (unsigned*)d_ws;         // 4*64 ordered-uint maxima

  pepair_init_amax<<<1, 256, 0, stream>>>(amax);
  pepair_pass1<<<32, 256, 0, stream>>>(x, W, amax);          // 32 blocks * 8 waves * 16 rows = 4096
  pepair_pass2<<<32, 256, 0, stream>>>(x, W, bias, amax, out);
}